// GSKANNet_33243046871156
// MI455X (gfx1250) — compile-verified
//
#include <hip/hip_runtime.h>

// GS-KAN fused layer for MI455X (gfx1250, wave32).
//
// out[b,q] = sum_p PReLU(x[b,p]+eps[q]) * Lam[p,q]
//          = c1*( (X.Lam)[b,q] + eps[q]*colsum(Lam)[q] ) + c2*sum_p |x[b,p]+eps[q]|*Lam[p,q]
// with c1=(1+a)/2, c2=(1-a)/2.
// Linear part -> V_WMMA_F32_16X16X4_F32.
// Abs part    -> 4x v_pk_add_f32 + 8x scalar v_fma_f32 with |.| source modifier
//                (inline asm pins the scalar form; VOP3P pk_fma has no ABS and
//                would cost 8 extra v_and_b32 per k).
// Per-k LDS loads (NOT group-batched: batching made the scheduler hoist loads
// across k4 groups, pushing VGPR use past 256 -> s_set_vgpr_msb + spills).
// Global->LDS software-pipelined through registers to hide HBM/L2 latency.

typedef __attribute__((ext_vector_type(2))) float v2f;
typedef __attribute__((ext_vector_type(8))) float v8f;

#define KC 32  // K chunk through LDS

// acc += |s| * lam  — scalar VOP3 FMA, abs folds into a free source modifier.
__device__ __forceinline__ void fma_abs(float& acc, float s, float lam) {
    asm("v_fma_f32 %0, |%1|, %2, %0" : "+v"(acc) : "v"(s), "v"(lam));
}

template<int N_IN, int N_OUT>
__global__ __launch_bounds__(256)
void gskan_layer(const float* __restrict__ X,    // [B, N_IN]
                 const float* __restrict__ Lam,  // [N_IN, N_OUT]
                 const float* __restrict__ Eps,  // [N_OUT]
                 const float* __restrict__ Ap,   // scalar a
                 float* __restrict__ Out)        // [B, N_OUT]
{
    // sX: transposed x chunk [k][b], padded to 36 floats/row (16B-aligned rows).
    // sL: Lam chunk [k][q].
    __shared__ float sX[KC][36];
    __shared__ float sL[KC][64];

    const int t    = threadIdx.x;
    const int wave = t >> 5;        // 0..7
    const int lane = t & 31;
    const int hi   = lane >> 4;     // 0/1 (upper lane half)
    const int l15  = lane & 15;
    const int wb   = wave >> 2;     // 0..1 : 16-row b sub-tile
    const int wq   = wave & 3;      // 0..3 : 16-col q sub-tile

    const int qBlock = blockIdx.x * 64;
    const int bBlock = blockIdx.y * 32;

    const int   q    = qBlock + wq * 16 + l15;
    const float epsq = Eps[q];
    const v2f   e2   = {epsq, epsq};
    const float a    = Ap[0];
    const float c1   = 0.5f * (1.0f + a);
    const float c2   = 0.5f * (1.0f - a);

    v8f accL = {0.f, 0.f, 0.f, 0.f, 0.f, 0.f, 0.f, 0.f};  // WMMA C (linear part)
    float accA[8];                                         // abs part, same (b,q) map
#pragma unroll
    for (int r = 0; r < 8; ++r) accA[r] = 0.f;
    float sumL = 0.f;                                      // colsum(Lam)[q]

    // cooperative-load index maps (256 threads)
    const int lb  = t >> 3;          // 0..31 : b row for X
    const int lk4 = (t & 7) << 2;    // 0..28 : k quad for X
    const int lkL = t >> 4;          // 0..15 : k row for Lam
    const int lq4 = (t & 15) << 2;   // 0..60 : q quad for Lam

    // ---- software pipeline: prefetch chunk 0 into registers ----
    float4 xv4 = *(const float4*)&X[(bBlock + lb) * N_IN + lk4];
    float4 l4a = *(const float4*)&Lam[(lkL)      * N_OUT + qBlock + lq4];
    float4 l4b = *(const float4*)&Lam[(lkL + 16) * N_OUT + qBlock + lq4];

    for (int k0 = 0; k0 < N_IN; k0 += KC) {
        __syncthreads();  // previous chunk fully consumed
        sX[lk4 + 0][lb] = xv4.x;   // transpose into [k][b]
        sX[lk4 + 1][lb] = xv4.y;
        sX[lk4 + 2][lb] = xv4.z;
        sX[lk4 + 3][lb] = xv4.w;
        *(float4*)&sL[lkL     ][lq4] = l4a;
        *(float4*)&sL[lkL + 16][lq4] = l4b;
        __syncthreads();

        // issue next chunk's global loads now; s_wait lands at next ds_store,
        // hidden behind this chunk's compute. Wave-uniform select, no branch.
        const int kn = (k0 + KC < N_IN) ? (k0 + KC) : 0;
        xv4 = *(const float4*)&X[(bBlock + lb) * N_IN + kn + lk4];
        l4a = *(const float4*)&Lam[(kn + lkL)      * N_OUT + qBlock + lq4];
        l4b = *(const float4*)&Lam[(kn + lkL + 16) * N_OUT + qBlock + lq4];

#pragma unroll
        for (int k4 = 0; k4 < KC; k4 += 4) {
            // ---- linear part: 16x16x4 f32 WMMA ----
            // A[M][K]: M = l15, K = v + 2*hi ; B[K][N]: N = l15, K = v + 2*hi
            v2f Am, Bm;
            Am.x = sX[k4 + 2 * hi + 0][wb * 16 + l15];
            Am.y = sX[k4 + 2 * hi + 1][wb * 16 + l15];
            Bm.x = sL[k4 + 2 * hi + 0][wq * 16 + l15];
            Bm.y = sL[k4 + 2 * hi + 1][wq * 16 + l15];
            accL = __builtin_amdgcn_wmma_f32_16x16x4_f32(
                       false, Am, false, Bm, (short)0, accL, false, false);

            // ---- abs part: acc[r] += |x[b,p]+eps[q]| * Lam[p,q] ----
#pragma unroll
            for (int kk = 0; kk < 4; ++kk) {
                const int   k    = k4 + kk;
                const float lamv = sL[k][wq * 16 + l15];
                sumL += lamv;  // colsum(Lam) for free
                const float4 xa = *(const float4*)&sX[k][wb * 16 + hi * 8];
                const float4 xb = *(const float4*)&sX[k][wb * 16 + hi * 8 + 4];
                // 4x v_pk_add_f32 (x pairs arrive even-aligned from ds_load_b128)
                v2f s01 = (v2f){xa.x, xa.y} + e2;
                v2f s23 = (v2f){xa.z, xa.w} + e2;
                v2f s45 = (v2f){xb.x, xb.y} + e2;
                v2f s67 = (v2f){xb.z, xb.w} + e2;
                // 8x scalar v_fma_f32 with free |.| modifier, independent chains
                fma_abs(accA[0], s01.x, lamv);
                fma_abs(accA[1], s01.y, lamv);
                fma_abs(accA[2], s23.x, lamv);
                fma_abs(accA[3], s23.y, lamv);
                fma_abs(accA[4], s45.x, lamv);
                fma_abs(accA[5], s45.y, lamv);
                fma_abs(accA[6], s67.x, lamv);
                fma_abs(accA[7], s67.y, lamv);
            }
        }
    }

    // combine (accL layout: M = r + 8*hi, N = l15 — matches accA mapping) and store
    const float lin_eps = epsq * sumL;
#pragma unroll
    for (int r = 0; r < 8; ++r) {
        const int b = bBlock + wb * 16 + hi * 8 + r;
        Out[b * N_OUT + q] = c1 * (accL[r] + lin_eps) + c2 * accA[r];
    }
}

extern "C" void kernel_launch(void* const* d_in, const int* in_sizes, int n_in,
                              void* d_out, int out_size, void* d_ws, size_t ws_size,
                              hipStream_t stream) {
    // inputs: x, (Lambda, eps, a) x 4
    const float* x   = (const float*)d_in[0];
    const float* L0  = (const float*)d_in[1];
    const float* e0  = (const float*)d_in[2];
    const float* a0  = (const float*)d_in[3];
    const float* L1  = (const float*)d_in[4];
    const float* e1  = (const float*)d_in[5];
    const float* a1  = (const float*)d_in[6];
    const float* L2  = (const float*)d_in[7];
    const float* e2  = (const float*)d_in[8];
    const float* a2  = (const float*)d_in[9];
    const float* L3  = (const float*)d_in[10];
    const float* e3  = (const float*)d_in[11];
    const float* a3  = (const float*)d_in[12];

    const int B = 1024;  // in_sizes[0] / 256
    float* buf0 = (float*)d_ws;             // [1024 x 512]
    float* buf1 = buf0 + (size_t)B * 512;   // [1024 x 512]
    float* out  = (float*)d_out;            // [1024 x 256]

    dim3 blk(256);
    // grid = (n_out/64, B/32)
    gskan_layer<256, 512><<<dim3(512 / 64, B / 32), blk, 0, stream>>>(x,    L0, e0, a0, buf0);
    gskan_layer<512, 512><<<dim3(512 / 64, B / 32), blk, 0, stream>>>(buf0, L1, e1, a1, buf1);
    gskan_layer<512, 512><<<dim3(512 / 64, B / 32), blk, 0, stream>>>(buf1, L2, e2, a2, buf0);
    gskan_layer<512, 256><<<dim3(256 / 64, B / 32), blk, 0, stream>>>(buf0, L3, e3, a3, out);
}